// SAGE_MLC_32478542692724
// MI455X (gfx1250) — compile-verified
//
#include <hip/hip_runtime.h>

typedef float v2f __attribute__((ext_vector_type(2)));
typedef float v8f __attribute__((ext_vector_type(8)));

#define N_NODES_C 10000
#define F_IN_C    128
#define HIDDEN_C  128
#define N_CLS_C   64
#define LDSTRIDE  132   // 128 + 4 pad: banks = (4*M + k) % 64 -> conflict-free readback
#define WPB       4     // waves (node tiles) per block in the GEMM kernel

// ---------------------------------------------------------------------------
// Kernel 1: edge-parallel mean-aggregation numerator/denominator.
// One wave per edge: 32 lanes x float4 = 128 features, coalesced 512B row
// read of x[src] (L2-resident), global_atomic_add_f32 into agg[dst].
// Edge indices are forced uniform (readfirstlane) -> scalar s_load path,
// and two edges are kept in flight per wave for memory-level parallelism.
// ---------------------------------------------------------------------------
__device__ __forceinline__ void process_edge(
    const float* __restrict__ x, const long long* __restrict__ ei,
    float* __restrict__ agg, float* __restrict__ cnt,
    int nEdges, int e, int lane)
{
    const int eu  = __builtin_amdgcn_readfirstlane(e);   // provably uniform -> s_load
    const int src = (int)ei[eu];
    const int dst = (int)ei[nEdges + eu];
    const float4 v = *(const float4*)(x + (size_t)src * F_IN_C + lane * 4);
    float* ap = agg + (size_t)dst * F_IN_C + lane * 4;
    atomicAdd(ap + 0, v.x);
    atomicAdd(ap + 1, v.y);
    atomicAdd(ap + 2, v.z);
    atomicAdd(ap + 3, v.w);
    if (lane == 0) atomicAdd(cnt + dst, 1.0f);
}

__global__ __launch_bounds__(256) void sage_aggregate(
    const float* __restrict__ x, const long long* __restrict__ ei,
    float* __restrict__ agg, float* __restrict__ cnt, int nEdges)
{
    const int lane   = threadIdx.x & 31;
    const int wave   = (int)((blockIdx.x * blockDim.x + threadIdx.x) >> 5);
    const int nWaves = (int)((gridDim.x * blockDim.x) >> 5);

    int e = wave;
    for (; e + nWaves < nEdges; e += 2 * nWaves) {
        process_edge(x, ei, agg, cnt, nEdges, e, lane);
        process_edge(x, ei, agg, cnt, nEdges, e + nWaves, lane);
    }
    if (e < nEdges)
        process_edge(x, ei, agg, cnt, nEdges, e, lane);
}

// ---------------------------------------------------------------------------
// Kernel 2: fused  h = relu(mean@Wl^T + b_l + x@Wr^T);  out = h@Wfc^T + b_fc
// using V_WMMA_F32_16X16X4_F32 (full fp32 precision). One wave per 16-node
// tile; h transposed C-layout -> A-layout through a padded per-wave LDS tile.
// ---------------------------------------------------------------------------
__global__ __launch_bounds__(32 * WPB) void sage_mlp(
    const float* __restrict__ x, const float* __restrict__ agg,
    const float* __restrict__ cnt,
    const float* __restrict__ Wl, const float* __restrict__ bl,
    const float* __restrict__ Wr,
    const float* __restrict__ Wfc, const float* __restrict__ bfc,
    float* __restrict__ out)
{
    __shared__ float hbuf[WPB][16 * LDSTRIDE];   // per-wave (16 x 132) fp32 tile

    const int wave = threadIdx.x >> 5;
    const int lane = threadIdx.x & 31;
    const int hi   = lane >> 4;        // half-wave select
    const int lrow = lane & 15;        // A-row (M) / B-col (N) for this lane
    const int kofs = hi * 2;           // per-lane K offset within a k0 group

    const int nTiles = N_NODES_C / 16; // 625
    for (int tile = blockIdx.x * WPB + wave; tile < nTiles;
         tile += gridDim.x * WPB) {
        const int n0   = tile * 16;
        const int arow = n0 + lrow;
        const float scale = 1.0f / fmaxf(cnt[arow], 1.0f);

        const float* aggRow = agg + (size_t)arow * F_IN_C + kofs;
        const float* xRow   = x   + (size_t)arow * F_IN_C + kofs;

        // ----- phase A: h accumulation over K=128 (two fp32 GEMMs fused) ----
        v8f acc[8] = {};
        for (int k0 = 0; k0 < F_IN_C; k0 += 4) {
            v2f aMean, aX;
            aMean.x = aggRow[k0]     * scale;
            aMean.y = aggRow[k0 + 1] * scale;
            aX.x    = xRow[k0];
            aX.y    = xRow[k0 + 1];
            #pragma unroll
            for (int c8 = 0; c8 < 8; ++c8) {
                const int nc = c8 * 16 + lrow;
                const float* wl = Wl + (size_t)nc * F_IN_C + k0 + kofs;
                v2f bL; bL.x = wl[0]; bL.y = wl[1];
                acc[c8] = __builtin_amdgcn_wmma_f32_16x16x4_f32(
                    false, aMean, false, bL, (short)0, acc[c8], false, false);
                const float* wr = Wr + (size_t)nc * F_IN_C + k0 + kofs;
                v2f bR; bR.x = wr[0]; bR.y = wr[1];
                acc[c8] = __builtin_amdgcn_wmma_f32_16x16x4_f32(
                    false, aX, false, bR, (short)0, acc[c8], false, false);
            }
        }

        // ----- bias + ReLU + stage h (C-layout -> LDS row-major) -----------
        #pragma unroll
        for (int c8 = 0; c8 < 8; ++c8) {
            const float bias = bl[c8 * 16 + lrow];
            #pragma unroll
            for (int r = 0; r < 8; ++r) {
                float v = acc[c8][r] + bias;
                v = fmaxf(v, 0.0f);
                hbuf[wave][(r + hi * 8) * LDSTRIDE + c8 * 16 + lrow] = v;
            }
        }
        asm volatile("s_wait_dscnt 0" ::: "memory");  // LDS transpose fence (wave-local)

        // ----- phase B: out = h @ Wfc^T + b_fc ------------------------------
        v8f oacc[4] = {};
        const float* hRow = &hbuf[wave][lrow * LDSTRIDE + kofs];
        for (int k0 = 0; k0 < HIDDEN_C; k0 += 4) {
            v2f aH; aH.x = hRow[k0]; aH.y = hRow[k0 + 1];
            #pragma unroll
            for (int c4 = 0; c4 < 4; ++c4) {
                const int nc = c4 * 16 + lrow;
                const float* wp = Wfc + (size_t)nc * HIDDEN_C + k0 + kofs;
                v2f b2; b2.x = wp[0]; b2.y = wp[1];
                oacc[c4] = __builtin_amdgcn_wmma_f32_16x16x4_f32(
                    false, aH, false, b2, (short)0, oacc[c4], false, false);
            }
        }

        #pragma unroll
        for (int c4 = 0; c4 < 4; ++c4) {
            const float bias = bfc[c4 * 16 + lrow];
            #pragma unroll
            for (int r = 0; r < 8; ++r) {
                out[(size_t)(n0 + r + hi * 8) * N_CLS_C + c4 * 16 + lrow] =
                    oacc[c4][r] + bias;
            }
        }
    }
}

// ---------------------------------------------------------------------------
extern "C" void kernel_launch(void* const* d_in, const int* in_sizes, int n_in,
                              void* d_out, int out_size, void* d_ws, size_t ws_size,
                              hipStream_t stream)
{
    const float*     x   = (const float*)d_in[0];
    const long long* ei  = (const long long*)d_in[1];  // int64 edge_index [2, E]
    const float*     Wl  = (const float*)d_in[2];
    const float*     bl  = (const float*)d_in[3];
    const float*     Wr  = (const float*)d_in[4];
    const float*     Wfc = (const float*)d_in[5];
    const float*     bfc = (const float*)d_in[6];
    float*           out = (float*)d_out;

    const int nEdges = in_sizes[1] / 2;

    float* agg = (float*)d_ws;                         // [N_NODES, F_IN]
    float* cnt = agg + (size_t)N_NODES_C * F_IN_C;     // [N_NODES]

    // zero accumulators every call (graph-capture-safe memset node)
    hipMemsetAsync(d_ws, 0,
                   ((size_t)N_NODES_C * F_IN_C + N_NODES_C) * sizeof(float),
                   stream);

    sage_aggregate<<<1024, 256, 0, stream>>>(x, ei, agg, cnt, nEdges);

    const int nTiles = N_NODES_C / 16;                 // 625
    const int grid2  = (nTiles + WPB - 1) / WPB;       // 157 blocks, 4 waves each
    sage_mlp<<<grid2, 32 * WPB, 0, stream>>>(x, agg, cnt, Wl, bl, Wr, Wfc, bfc, out);
}